// myGraph_gat_25074019074260
// MI455X (gfx1250) — compile-verified
//
#include <hip/hip_runtime.h>

#define NODES  50000
#define EDGES  800000
#define INF_   256
#define HIDF   128
#define OUTF   8
#define NGRAPH 256

typedef __attribute__((ext_vector_type(2))) float v2f;
typedef __attribute__((ext_vector_type(8))) float v8f;

// Order-preserving float<->uint encoding so atomicMax(unsigned) == float max.
__device__ __forceinline__ unsigned enc_f(float f) {
    unsigned b = __float_as_uint(f);
    return (b & 0x80000000u) ? ~b : (b | 0x80000000u);
}
__device__ __forceinline__ float dec_f(unsigned k) {
    return (k & 0x80000000u) ? __uint_as_float(k & 0x7fffffffu)
                             : __uint_as_float(~k);
}

// ---------------------------------------------------------------------------
// C[M x 128] = A[M x K] @ B[K x 128], fp32 WMMA 16x16x4.
// Block = 128 threads (4 waves). blockIdx.x = M-tile (16 rows).
// Wave w computes a 16x32 strip: two 16x16 N-tiles at columns [32w, 32w+32),
// sharing one A fragment per K-step (2 WMMAs per A load).
// A frag (16x4): lanes 0-15 hold K={k,k+1}, lanes 16-31 hold K={k+2,k+3} (M=lane%16).
// B frag (4x16): lanes 0-15 hold K={k,k+1}, lanes 16-31 hold K={k+2,k+3} (N=lane%16).
// C/D (16x16 f32): VGPR r, lanes 0-15 -> M=r, lanes 16-31 -> M=r+8, N=lane%16.
// ---------------------------------------------------------------------------
template <int KDIM>
__global__ __launch_bounds__(128) void wmma_gemm_n128(
    const float* __restrict__ A, const float* __restrict__ B,
    float* __restrict__ C) {
    const int lane = threadIdx.x & 31;
    const int wave = threadIdx.x >> 5;
    const int m0   = blockIdx.x * 16;
    const int n0   = wave * 32;
    const int lhi  = lane >> 4;   // 0 or 1 (selects K pair)
    const int llo  = lane & 15;   // M (for A) / N (for B)

    const float* arow = A + (size_t)(m0 + llo) * KDIM + 2 * lhi;
    const float* bp0  = B + (size_t)(2 * lhi) * HIDF + n0 + llo;
    const float* bp1  = bp0 + 16;

    v8f c0 = {};
    v8f c1 = {};
#pragma unroll 4
    for (int k = 0; k < KDIM; k += 4) {
        v2f a, b0, b1;
        a.x  = arow[k];
        a.y  = arow[k + 1];
        b0.x = bp0[(size_t)k * HIDF];
        b0.y = bp0[(size_t)k * HIDF + HIDF];
        b1.x = bp1[(size_t)k * HIDF];
        b1.y = bp1[(size_t)k * HIDF + HIDF];
        c0 = __builtin_amdgcn_wmma_f32_16x16x4_f32(false, a, false, b0,
                                                   (short)0, c0, false, false);
        c1 = __builtin_amdgcn_wmma_f32_16x16x4_f32(false, a, false, b1,
                                                   (short)0, c1, false, false);
    }

    float* crow = C + (size_t)(m0 + 8 * lhi) * HIDF + n0 + llo;
#pragma unroll
    for (int r = 0; r < 8; ++r) {
        crow[(size_t)r * HIDF]      = c0[r];
        crow[(size_t)r * HIDF + 16] = c1[r];
    }
}

// alpha_s[n] = h[n,:]·a_src ; alpha_d[n] = h[n,:]·a_dst  (one wave per node)
__global__ __launch_bounds__(256) void alphas_k(
    const float* __restrict__ h, const float* __restrict__ asrc,
    const float* __restrict__ adst, float* __restrict__ out_s,
    float* __restrict__ out_d) {
    const int lane = threadIdx.x & 31;
    const int node = blockIdx.x * 8 + (threadIdx.x >> 5);
    if (node >= NODES) return;
    const float4 hv = reinterpret_cast<const float4*>(h + (size_t)node * HIDF)[lane];
    const float4 sv = reinterpret_cast<const float4*>(asrc)[lane];
    const float4 dv = reinterpret_cast<const float4*>(adst)[lane];
    float ps = hv.x * sv.x + hv.y * sv.y + hv.z * sv.z + hv.w * sv.w;
    float pd = hv.x * dv.x + hv.y * dv.y + hv.z * dv.z + hv.w * dv.w;
#pragma unroll
    for (int off = 16; off > 0; off >>= 1) {
        ps += __shfl_down(ps, off, 32);
        pd += __shfl_down(pd, off, 32);
    }
    if (lane == 0) { out_s[node] = ps; out_d[node] = pd; }
}

// Pass 1: logit = leaky_relu(as[src]+ad[dst]); store; segment-max via atomicMax.
__global__ void edge_logit_k(const int* __restrict__ src, const int* __restrict__ dst,
                             const float* __restrict__ as_, const float* __restrict__ ad_,
                             float* __restrict__ ebuf, unsigned* __restrict__ mx) {
    int e = blockIdx.x * blockDim.x + threadIdx.x;
    if (e >= EDGES) return;
    const int s = src[e], d = dst[e];
    float v = as_[s] + ad_[d];
    v = (v > 0.0f) ? v : 0.2f * v;   // NEG_SLOPE = 0.2
    ebuf[e] = v;
    atomicMax(&mx[d], enc_f(v));
}

// Pass 2 (fused exp + scatter): ex = exp(logit - max[dst]);
//   ssum[dst] += ex;  agg[dst,:] += ex * h[src,:]   (one wave per edge)
// Normalization by ssum is deferred to the node-level epilogue.
__global__ __launch_bounds__(256) void edge_exp_scatter_k(
    const int* __restrict__ src, const int* __restrict__ dst,
    const float* __restrict__ ebuf, const unsigned* __restrict__ mx,
    float* __restrict__ ssum, const float* __restrict__ h,
    float* __restrict__ agg) {
    const int lane = threadIdx.x & 31;
    const int e = blockIdx.x * 8 + (threadIdx.x >> 5);
    if (e >= EDGES) return;
    const int s = src[e], d = dst[e];
    const float ex = __expf(ebuf[e] - dec_f(mx[d]));
    if (lane == 0) atomicAdd(&ssum[d], ex);
    const float4 hv = reinterpret_cast<const float4*>(h + (size_t)s * HIDF)[lane];
    float* o = agg + (size_t)d * HIDF + lane * 4;
    atomicAdd(o + 0, ex * hv.x);
    atomicAdd(o + 1, ex * hv.y);
    atomicAdd(o + 2, ex * hv.z);
    atomicAdd(o + 3, ex * hv.w);
}

// Epilogue: q = (ssum>0 ? q/ssum : 0) + bias, optional ReLU. In-place on Q.
__global__ void bias_act_k(float* __restrict__ q, const float* __restrict__ ssum,
                           const float* __restrict__ bias, int relu) {
    size_t i = (size_t)blockIdx.x * blockDim.x + threadIdx.x;
    if (i >= (size_t)NODES * HIDF) return;
    const float s = ssum[i >> 7];          // i / HIDF
    float v = (s > 0.0f) ? q[i] / s : 0.0f;
    v += bias[i & (HIDF - 1)];
    if (relu) v = fmaxf(v, 0.0f);
    q[i] = v;
}

// Pooled sums + counts (one wave per node)
__global__ __launch_bounds__(256) void pool_k(
    const float* __restrict__ h, const int* __restrict__ batch,
    float* __restrict__ pooled, float* __restrict__ cnt) {
    const int lane = threadIdx.x & 31;
    const int node = blockIdx.x * 8 + (threadIdx.x >> 5);
    if (node >= NODES) return;
    const int g = batch[node];
    const float4 hv = reinterpret_cast<const float4*>(h + (size_t)node * HIDF)[lane];
    float* p = pooled + (size_t)g * HIDF + lane * 4;
    atomicAdd(p + 0, hv.x);
    atomicAdd(p + 1, hv.y);
    atomicAdd(p + 2, hv.z);
    atomicAdd(p + 3, hv.w);
    if (lane == 0) atomicAdd(&cnt[g], 1.0f);
}

// out[g,o] = (sum[g,:]@lin_w)[o] / max(cnt[g],1) + lin_b[o]
__global__ void final_k(const float* __restrict__ pooled, const float* __restrict__ cnt,
                        const float* __restrict__ lw, const float* __restrict__ lb,
                        float* __restrict__ out) {
    int t = blockIdx.x * blockDim.x + threadIdx.x;
    if (t >= NGRAPH * OUTF) return;
    int g = t / OUTF, o = t % OUTF;
    float acc = 0.0f;
#pragma unroll 8
    for (int k = 0; k < HIDF; ++k) acc += pooled[g * HIDF + k] * lw[k * OUTF + o];
    float c = cnt[g];
    c = (c > 1.0f) ? c : 1.0f;
    out[t] = acc / c + lb[o];
}

extern "C" void kernel_launch(void* const* d_in, const int* in_sizes, int n_in,
                              void* d_out, int out_size, void* d_ws, size_t ws_size,
                              hipStream_t stream) {
    (void)in_sizes; (void)n_in; (void)out_size; (void)ws_size;
    const float* x          = (const float*)d_in[0];
    const int*   edge_index = (const int*)d_in[1];
    // d_in[2] = edge_attr, unused (matches reference else-branch)
    const int*   batch      = (const int*)d_in[3];
    const float* W[3]    = {(const float*)d_in[4],  (const float*)d_in[8],  (const float*)d_in[12]};
    const float* Asrc[3] = {(const float*)d_in[5],  (const float*)d_in[9],  (const float*)d_in[13]};
    const float* Adst[3] = {(const float*)d_in[6],  (const float*)d_in[10], (const float*)d_in[14]};
    const float* Bias[3] = {(const float*)d_in[7],  (const float*)d_in[11], (const float*)d_in[15]};
    const float* lin_w = (const float*)d_in[16];
    const float* lin_b = (const float*)d_in[17];
    float* out = (float*)d_out;

    const int* src = edge_index;           // row 0
    const int* dst = edge_index + EDGES;   // row 1

    // Workspace carve-out (~55 MB)
    char* ws = (char*)d_ws;
    size_t off = 0;
    auto carve = [&](size_t bytes) -> void* {
        void* p = ws + off;
        off += (bytes + 255) & ~(size_t)255;
        return p;
    };
    float*    P      = (float*)carve(sizeof(float) * NODES * HIDF);   // h = in @ W
    float*    Q      = (float*)carve(sizeof(float) * NODES * HIDF);   // aggregated / next input
    float*    as_    = (float*)carve(sizeof(float) * NODES);
    float*    ad_    = (float*)carve(sizeof(float) * NODES);
    float*    ssum   = (float*)carve(sizeof(float) * NODES);
    unsigned* mx     = (unsigned*)carve(sizeof(unsigned) * NODES);
    float*    ebuf   = (float*)carve(sizeof(float) * EDGES);
    float*    pooled = (float*)carve(sizeof(float) * NGRAPH * HIDF);
    float*    cnt    = (float*)carve(sizeof(float) * NGRAPH);

    for (int l = 0; l < 3; ++l) {
        const float* in = (l == 0) ? x : Q;
        if (l == 0)
            wmma_gemm_n128<INF_><<<NODES / 16, 128, 0, stream>>>(in, W[l], P);
        else
            wmma_gemm_n128<HIDF><<<NODES / 16, 128, 0, stream>>>(in, W[l], P);
        alphas_k<<<(NODES + 7) / 8, 256, 0, stream>>>(P, Asrc[l], Adst[l], as_, ad_);
        hipMemsetAsync(Q, 0, sizeof(float) * NODES * HIDF, stream);
        hipMemsetAsync(ssum, 0, sizeof(float) * NODES, stream);
        hipMemsetAsync(mx, 0, sizeof(unsigned) * NODES, stream);   // enc key 0 == -inf
        edge_logit_k<<<(EDGES + 255) / 256, 256, 0, stream>>>(src, dst, as_, ad_, ebuf, mx);
        edge_exp_scatter_k<<<(EDGES + 7) / 8, 256, 0, stream>>>(src, dst, ebuf, mx, ssum, P, Q);
        bias_act_k<<<((size_t)NODES * HIDF + 255) / 256, 256, 0, stream>>>(Q, ssum, Bias[l], l < 2 ? 1 : 0);
    }

    hipMemsetAsync(pooled, 0, sizeof(float) * NGRAPH * HIDF, stream);
    hipMemsetAsync(cnt, 0, sizeof(float) * NGRAPH, stream);
    pool_k<<<(NODES + 7) / 8, 256, 0, stream>>>(Q, batch, pooled, cnt);
    final_k<<<(NGRAPH * OUTF + 255) / 256, 256, 0, stream>>>(pooled, cnt, lin_w, lin_b, out);
}